// SpectralGNN_53815940218921
// MI455X (gfx1250) — compile-verified
//
#include <hip/hip_runtime.h>

typedef float v2f __attribute__((ext_vector_type(2)));
typedef float v8f __attribute__((ext_vector_type(8)));

#define HID 64
#define INCH 16
#define OUTCH 4

// ---------------- helpers ----------------
__device__ inline void atomAddF(float* p, float v) {
    // relaxed, device-scope, no-return -> global_atomic_add_f32
    __hip_atomic_fetch_add(p, v, __ATOMIC_RELAXED, __HIP_MEMORY_SCOPE_AGENT);
}

// Accumulate C(16x64 tile) += scale*A(16xKD) * W(KDx64), W staged in LDS.
// A rows are dense with stride KD. Uses V_WMMA_F32_16X16X4_F32 (full f32).
template <int KD>
__device__ inline void tile_mma_f32(const float* __restrict__ A,
                                    const float* __restrict__ Wlds,
                                    float scale, v8f c[4], int lane) {
    const int m     = lane & 15;
    const int khalf = (lane >> 4) * 2;  // 0 for lanes 0-15, 2 for lanes 16-31
#pragma unroll
    for (int k0 = 0; k0 < KD; k0 += 4) {
        v2f a;
        const float* ap = A + m * KD + k0 + khalf;
        a.x = ap[0] * scale;
        a.y = ap[1] * scale;
#pragma unroll
        for (int j = 0; j < 4; ++j) {
            v2f b;
            const float* bp = Wlds + (k0 + khalf) * HID + (lane & 15) + j * 16;
            b.x = bp[0];
            b.y = bp[HID];
            c[j] = __builtin_amdgcn_wmma_f32_16x16x4_f32(
                false, a, false, b, (short)0, c[j], false, false);
        }
    }
}

// ---------------- degree / edge weights ----------------
__global__ void deg_kernel(const int* __restrict__ src, float* __restrict__ deg, int E) {
    int e = blockIdx.x * blockDim.x + threadIdx.x;
    if (e < E) atomAddF(&deg[src[e]], 1.0f);
}

__global__ void edgew_kernel(const int* __restrict__ src, const int* __restrict__ dst,
                             const float* __restrict__ deg, float* __restrict__ w, int E) {
    int e = blockIdx.x * blockDim.x + threadIdx.x;
    if (e >= E) return;
    float ds = deg[src[e]], dd = deg[dst[e]];
    float is = ds > 0.0f ? rsqrtf(fmaxf(ds, 1.0f)) : 0.0f;
    float id = dd > 0.0f ? rsqrtf(fmaxf(dd, 1.0f)) : 0.0f;
    w[e] = -is * id;
}

// ---------------- h = x @ W_in + b_in  (WMMA f32, K=16) ----------------
__global__ void gemm_in_kernel(const float* __restrict__ x, const float* __restrict__ Win,
                               const float* __restrict__ bin, float* __restrict__ h, int n) {
    __shared__ float Wl[INCH * HID];  // 4KB
    __shared__ float Bl[HID];
    int t = threadIdx.x;
    for (int i = t; i < INCH * HID / 4; i += blockDim.x)
        ((float4*)Wl)[i] = ((const float4*)Win)[i];
    if (t < HID) Bl[t] = bin[t];
    __syncthreads();

    int lane = t & 31;
    int rowblk = blockIdx.x * (blockDim.x >> 5) + (t >> 5);
    int row0 = rowblk * 16;
    if (row0 >= n) return;

    v8f c[4] = {};
    tile_mma_f32<INCH>(x + (size_t)row0 * INCH, Wl, 1.0f, c, lane);

    int m0 = (lane >> 4) * 8;
    int col = lane & 15;
#pragma unroll
    for (int j = 0; j < 4; ++j)
#pragma unroll
        for (int r = 0; r < 8; ++r)
            h[(size_t)(row0 + m0 + r) * HID + col + j * 16] = c[j][r] + Bl[col + j * 16];
}

// ---------------- tx2 = -0.5 * h ----------------
__global__ void scaleinit_kernel(const float* __restrict__ h, float* __restrict__ tx2, int n4) {
    int i = blockIdx.x * blockDim.x + threadIdx.x;
    if (i >= n4) return;
    float4 v = ((const float4*)h)[i];
    ((float4*)tx2)[i] = make_float4(-0.5f * v.x, -0.5f * v.y, -0.5f * v.z, -0.5f * v.w);
}

// ------ acc[dst] += w * hin[src]  (32 edges per wave, shfl broadcast) ------
// Lane L loads metadata for edge base+L (fully coalesced), then the wave
// iterates the 32 edges: all lanes gather one 64ch row (float2/lane) and
// issue 2 hardware f32 atomics each.
__global__ void prop_kernel(const int* __restrict__ src, const int* __restrict__ dst,
                            const float* __restrict__ w, const float* __restrict__ hin,
                            float* __restrict__ acc, int E) {
    int wid  = (blockIdx.x * blockDim.x + threadIdx.x) >> 5;
    int lane = threadIdx.x & 31;
    int base = wid * 32;
    if (base >= E) return;
    int cnt = E - base;
    if (cnt > 32) cnt = 32;

    int s = 0, d = 0;
    float we = 0.0f;
    if (lane < cnt) {
        s  = src[base + lane];
        d  = dst[base + lane];
        we = w[base + lane];
    }

    for (int i = 0; i < cnt; ++i) {
        int   si = __shfl(s, i, 32);
        int   di = __shfl(d, i, 32);
        float wi = __shfl(we, i, 32);
        float2 v = *(const float2*)(hin + (size_t)si * HID + lane * 2);
        float* ap = acc + (size_t)di * HID + lane * 2;
        atomAddF(ap + 0, wi * v.x);
        atomAddF(ap + 1, wi * v.y);
    }
}

// ------- out = h@W0 + tx1@W1 + 2*tx2@W2   (fused 3-input WMMA GEMM) -------
__global__ void gemm3_kernel(const float* __restrict__ h, const float* __restrict__ tx1,
                             const float* __restrict__ tx2, const float* __restrict__ W3,
                             float* __restrict__ out, int n) {
    __shared__ float Wl[3 * HID * HID];  // 48KB of 320KB WGP LDS
    int t = threadIdx.x;
    for (int i = t; i < 3 * HID * HID / 4; i += blockDim.x)
        ((float4*)Wl)[i] = ((const float4*)W3)[i];
    __syncthreads();

    int lane = t & 31;
    int rowblk = blockIdx.x * (blockDim.x >> 5) + (t >> 5);
    int row0 = rowblk * 16;
    if (row0 >= n) return;

    v8f c[4] = {};
    tile_mma_f32<HID>(h   + (size_t)row0 * HID, Wl,                 1.0f, c, lane);
    tile_mma_f32<HID>(tx1 + (size_t)row0 * HID, Wl + HID * HID,     1.0f, c, lane);
    tile_mma_f32<HID>(tx2 + (size_t)row0 * HID, Wl + 2 * HID * HID, 2.0f, c, lane);

    int m0 = (lane >> 4) * 8;
    int col = lane & 15;
#pragma unroll
    for (int j = 0; j < 4; ++j)
#pragma unroll
        for (int r = 0; r < 8; ++r)
            out[(size_t)(row0 + m0 + r) * HID + col + j * 16] = c[j][r];
}

// ---- h = LayerNorm(relu(out + bias) + h) * g + b   (one wave per node) ----
__global__ void epilogue_kernel(const float* __restrict__ out, const float* __restrict__ bias,
                                const float* __restrict__ g, const float* __restrict__ b,
                                float* __restrict__ h, int n) {
    int wid = (blockIdx.x * blockDim.x + threadIdx.x) >> 5;
    int lane = threadIdx.x & 31;
    if (wid >= n) return;
    size_t base = (size_t)wid * HID + lane * 2;
    float2 o  = *(const float2*)(out + base);
    float2 bb = *(const float2*)(bias + lane * 2);
    float2 id = *(const float2*)(h + base);
    float a0 = fmaxf(o.x + bb.x, 0.0f) + id.x;
    float a1 = fmaxf(o.y + bb.y, 0.0f) + id.y;
    float s = a0 + a1;
#pragma unroll
    for (int m = 16; m >= 1; m >>= 1) s += __shfl_xor(s, m, 32);
    float mu = s * (1.0f / HID);
    float d0 = a0 - mu, d1 = a1 - mu;
    float v = d0 * d0 + d1 * d1;
#pragma unroll
    for (int m = 16; m >= 1; m >>= 1) v += __shfl_xor(v, m, 32);
    float rs = rsqrtf(v * (1.0f / HID) + 1e-5f);
    float2 gg = *(const float2*)(g + lane * 2);
    float2 lb = *(const float2*)(b + lane * 2);
    *(float2*)(h + base) = make_float2(d0 * rs * gg.x + lb.x, d1 * rs * gg.y + lb.y);
}

// ---------------- y = h @ W_out + b_out ----------------
__global__ void gemm_out_kernel(const float* __restrict__ h, const float* __restrict__ Wout,
                                const float* __restrict__ bout, float* __restrict__ y, int n) {
    __shared__ float Wl[HID * OUTCH];
    __shared__ float Bl[OUTCH];
    int t = threadIdx.x;
    if (t < HID * OUTCH) Wl[t] = Wout[t];
    if (t < OUTCH) Bl[t] = bout[t];
    __syncthreads();
    int i = blockIdx.x * blockDim.x + t;
    if (i >= n) return;
    float acc0 = Bl[0], acc1 = Bl[1], acc2 = Bl[2], acc3 = Bl[3];
    const float* hr = h + (size_t)i * HID;
#pragma unroll
    for (int k = 0; k < HID; ++k) {
        float hv = hr[k];
        acc0 += hv * Wl[k * OUTCH + 0];
        acc1 += hv * Wl[k * OUTCH + 1];
        acc2 += hv * Wl[k * OUTCH + 2];
        acc3 += hv * Wl[k * OUTCH + 3];
    }
    *(float4*)(y + (size_t)i * OUTCH) = make_float4(acc0, acc1, acc2, acc3);
}

// ---------------- host orchestration ----------------
extern "C" void kernel_launch(void* const* d_in, const int* in_sizes, int n_in,
                              void* d_out, int out_size, void* d_ws, size_t ws_size,
                              hipStream_t stream) {
    const float* x     = (const float*)d_in[0];
    const int*   eidx  = (const int*)d_in[1];
    const float* Win   = (const float*)d_in[3];
    const float* bin   = (const float*)d_in[4];
    const float* chebW = (const float*)d_in[5];
    const float* chebB = (const float*)d_in[6];
    const float* lng   = (const float*)d_in[7];
    const float* lnb   = (const float*)d_in[8];
    const float* Wout  = (const float*)d_in[9];
    const float* bout  = (const float*)d_in[10];
    float* y = (float*)d_out;

    const int N = in_sizes[0] / INCH;
    const int E = in_sizes[1] / 2;
    const int* src = eidx;
    const int* dst = eidx + E;

    // workspace layout (floats)
    float* ws  = (float*)d_ws;
    float* deg = ws;                 // N
    float* w   = deg + N;            // E
    float* h   = w + E;              // N*HID
    float* tx1 = h + (size_t)N * HID;
    float* tx2 = tx1 + (size_t)N * HID;
    float* out = tx2 + (size_t)N * HID;

    const int TB = 256;
    const int eBlocks    = (E + TB - 1) / TB;
    const int rowBlks    = (N + 15) / 16;
    const int gemmBlocks = (rowBlks + (TB / 32) - 1) / (TB / 32);
    const int edgesPerBlock = (TB / 32) * 32;  // 8 waves * 32 edges
    const int propBlocks = (E + edgesPerBlock - 1) / edgesPerBlock;
    const int nodeWaveBl = (N + (TB / 32) - 1) / (TB / 32);     // wave per node
    const int n4         = N * HID / 4;
    const int n4Blocks   = (n4 + TB - 1) / TB;
    const int nBlocks    = (N + TB - 1) / TB;

    // degree + symmetric-norm edge weights: w = -dinv[src]*dinv[dst]
    hipMemsetAsync(deg, 0, (size_t)N * sizeof(float), stream);
    deg_kernel<<<eBlocks, TB, 0, stream>>>(src, deg, E);
    edgew_kernel<<<eBlocks, TB, 0, stream>>>(src, dst, deg, w, E);

    // input projection
    gemm_in_kernel<<<gemmBlocks, TB, 0, stream>>>(x, Win, bin, h, N);

    for (int i = 0; i < 3; ++i) {
        // Tx1 = prop(h)
        hipMemsetAsync(tx1, 0, (size_t)N * HID * sizeof(float), stream);
        prop_kernel<<<propBlocks, TB, 0, stream>>>(src, dst, w, h, tx1, E);
        // tx2 = prop(Tx1) - 0.5*h   (so Tx2 = 2*tx2)
        scaleinit_kernel<<<n4Blocks, TB, 0, stream>>>(h, tx2, n4);
        prop_kernel<<<propBlocks, TB, 0, stream>>>(src, dst, w, tx1, tx2, E);
        // out = h@W0 + Tx1@W1 + 2*tx2@W2
        gemm3_kernel<<<gemmBlocks, TB, 0, stream>>>(h, tx1, tx2,
                                                    chebW + (size_t)i * 3 * HID * HID, out, N);
        // h = LN(relu(out + b) + h)
        epilogue_kernel<<<nodeWaveBl, TB, 0, stream>>>(out, chebB + i * HID,
                                                       lng + i * HID, lnb + i * HID, h, N);
    }

    gemm_out_kernel<<<nBlocks, TB, 0, stream>>>(h, Wout, bout, y, N);
}